// JointParagraphKGATClassifier_38156489458418
// MI455X (gfx1250) — compile-verified
//
#include <hip/hip_runtime.h>
#include <math.h>

typedef __attribute__((ext_vector_type(2))) float v2f;
typedef __attribute__((ext_vector_type(8))) float v8f;

#define NB 4
#define NS 16
#define NT 128
#define ND 768
#define NH 128

// Gaussian kernel constants: mu_k, and -0.5/sigma_k^2
__device__ __constant__ float c_MU[6] = {1.0f, 0.8f, 0.4f, 0.0f, -0.4f, -0.8f};
__device__ __constant__ float c_NC[6] = {-500000.0f, -50.0f, -50.0f, -50.0f, -50.0f, -50.0f};

__device__ __forceinline__ float waveSum(float v) {
  v += __shfl_xor(v, 16, 32);
  v += __shfl_xor(v, 8, 32);
  v += __shfl_xor(v, 4, 32);
  v += __shfl_xor(v, 2, 32);
  v += __shfl_xor(v, 1, 32);
  return v;
}

// ---------------------------------------------------------------------------
// K1: l2-normalize sentence_token_reps rows (8192) and claim_reps rows (512)
// ---------------------------------------------------------------------------
__global__ __launch_bounds__(256)
void norm_kernel(const float* __restrict__ reps, const float* __restrict__ claim,
                 float* __restrict__ rn, float* __restrict__ cn) {
  __shared__ float red[8];
  int blk = blockIdx.x;
  const float* src;
  float* dst;
  if (blk < NB * NS * NT) {
    src = reps + (size_t)blk * ND;
    dst = rn + (size_t)blk * ND;
  } else {
    int r = blk - NB * NS * NT;
    src = claim + (size_t)r * ND;
    dst = cn + (size_t)r * ND;
  }
  float ss = 0.0f;
  for (int d = threadIdx.x; d < ND; d += 256) {
    float x = src[d];
    ss += x * x;
  }
  ss = waveSum(ss);
  if ((threadIdx.x & 31) == 0) red[threadIdx.x >> 5] = ss;
  __syncthreads();
  float tot = 0.0f;
#pragma unroll
  for (int k = 0; k < 8; k++) tot += red[k];
  float inv = 1.0f / fmaxf(sqrtf(tot), 1e-6f);
  for (int d = threadIdx.x; d < ND; d += 256) dst[d] = src[d] * inv;
}

// ---------------------------------------------------------------------------
// K2: WMMA f32 Gram tile (128x128x768) per (b,i,j), fused Gaussian pooling,
//     log-clip, w_sel projection, softmax over t -> att[b,i,j,t]
// ---------------------------------------------------------------------------
#define KC 64
#define LDST 68  // padded LDS row stride (floats) -> conflict-free b64 frag reads

__global__ __launch_bounds__(256)
void pool_att_kernel(const float* __restrict__ rn,
                     const float* __restrict__ w_sel, const float* __restrict__ b_sel,
                     float* __restrict__ att) {
  __shared__ float As[NT * LDST];
  __shared__ float Bs[NT * LDST];
  __shared__ float sc[NT];
  __shared__ float red[8];

  int blk = blockIdx.x;               // (b*16 + i)*16 + j
  int b = blk >> 8;
  int i = (blk >> 4) & 15;
  int j = blk & 15;
  int tid = threadIdx.x;
  int wv = tid >> 5;
  int lane = tid & 31;
  int lg = lane & 15;   // lane within 16-group
  int hi = lane >> 4;   // 0 or 1

  const float* Abase = rn + ((size_t)(b * NS + i) * NT) * ND;
  const float* Bbase = rn + ((size_t)(b * NS + j) * NT) * ND;

  v8f acc[8];
#pragma unroll
  for (int c = 0; c < 8; c++)
#pragma unroll
    for (int e = 0; e < 8; e++) acc[c][e] = 0.0f;

  for (int kc = 0; kc < ND; kc += KC) {
    // stage A (rows of rn[b,i]) and B (rows of rn[b,j]) chunks into LDS
    for (int l = tid; l < NT * (KC / 4); l += 256) {
      int row = l >> 4;
      int c4 = l & 15;
      float4 av = *(const float4*)(Abase + (size_t)row * ND + kc + c4 * 4);
      float4 bv = *(const float4*)(Bbase + (size_t)row * ND + kc + c4 * 4);
      *(float4*)(&As[row * LDST + c4 * 4]) = av;
      *(float4*)(&Bs[row * LDST + c4 * 4]) = bv;
    }
    __syncthreads();

    for (int ks = 0; ks < KC / 4; ks++) {
      int kk = ks * 4 + hi * 2;
      const float* ap = &As[(wv * 16 + lg) * LDST + kk];
      v2f a;
      a.x = ap[0];
      a.y = ap[1];
#pragma unroll
      for (int c = 0; c < 8; c++) {
        const float* bp = &Bs[(c * 16 + lg) * LDST + kk];
        v2f bb;
        bb.x = bp[0];
        bb.y = bp[1];
        acc[c] = __builtin_amdgcn_wmma_f32_16x16x4_f32(
            false, a, false, bb, (short)0, acc[c], false, false);
      }
    }
    __syncthreads();
  }

  // --- fused epilogue: 6 Gaussian kernels, sum over u -------------------
  float pk[8][6];
#pragma unroll
  for (int r = 0; r < 8; r++)
#pragma unroll
    for (int k = 0; k < 6; k++) pk[r][k] = 0.0f;

#pragma unroll
  for (int c = 0; c < 8; c++) {
#pragma unroll
    for (int r = 0; r < 8; r++) {
      float s = acc[c][r];
#pragma unroll
      for (int k = 0; k < 6; k++) {
        float d = s - c_MU[k];
        pk[r][k] += expf(d * d * c_NC[k]);
      }
    }
  }
  // reduce over the 16 lanes of each half-wave (completes the sum over u)
#pragma unroll
  for (int r = 0; r < 8; r++) {
#pragma unroll
    for (int k = 0; k < 6; k++) {
      float v = pk[r][k];
      v += __shfl_xor(v, 1, 32);
      v += __shfl_xor(v, 2, 32);
      v += __shfl_xor(v, 4, 32);
      v += __shfl_xor(v, 8, 32);
      pk[r][k] = v;
    }
  }
  float wsel[6];
#pragma unroll
  for (int k = 0; k < 6; k++) wsel[k] = w_sel[k];
  float bsel = b_sel[0];
  if (lg == 0) {
#pragma unroll
    for (int r = 0; r < 8; r++) {
      float s = bsel;
#pragma unroll
      for (int k = 0; k < 6; k++) s += logf(fmaxf(pk[r][k], 1e-6f)) * wsel[k];
      sc[wv * 16 + hi * 8 + r] = s;  // t = 16*wv + hi*8 + r
    }
  }
  __syncthreads();

  // --- softmax over t (128 values) --------------------------------------
  float v = (tid < NT) ? sc[tid] : -3.0e38f;
  float m = v;
  m = fmaxf(m, __shfl_xor(m, 16, 32));
  m = fmaxf(m, __shfl_xor(m, 8, 32));
  m = fmaxf(m, __shfl_xor(m, 4, 32));
  m = fmaxf(m, __shfl_xor(m, 2, 32));
  m = fmaxf(m, __shfl_xor(m, 1, 32));
  if (lane == 0) red[wv] = m;
  __syncthreads();
  float gm = red[0];
#pragma unroll
  for (int k = 1; k < 8; k++) gm = fmaxf(gm, red[k]);
  __syncthreads();
  float e = (tid < NT) ? expf(v - gm) : 0.0f;
  float ssum = waveSum(e);
  if (lane == 0) red[wv] = ssum;
  __syncthreads();
  float tot = 0.0f;
#pragma unroll
  for (int k = 0; k < 8; k++) tot += red[k];
  if (tid < NT) att[(size_t)blk * NT + tid] = e / tot;
}

// ---------------------------------------------------------------------------
// K3: z_hat[b,i,j,d] = sum_t att[b,i,j,t] * reps[b,j,t,d]
// ---------------------------------------------------------------------------
__global__ __launch_bounds__(256)
void zhat_kernel(const float* __restrict__ reps, const float* __restrict__ att,
                 float* __restrict__ zhat) {
  __shared__ float a[NT];
  int blk = blockIdx.x;  // (b*16+i)*16+j
  int b = blk >> 8;
  int j = blk & 15;
  int tid = threadIdx.x;
  if (tid < NT) a[tid] = att[(size_t)blk * NT + tid];
  __syncthreads();
  const float* R = reps + ((size_t)(b * NS + j) * NT) * ND;
#pragma unroll
  for (int dd = 0; dd < 3; dd++) {
    int d = tid + dd * 256;
    float s = 0.0f;
    for (int t = 0; t < NT; t++) s += a[t] * R[(size_t)t * ND + d];
    zhat[(size_t)blk * ND + d] = s;
  }
}

// ---------------------------------------------------------------------------
// K4: g[b,i,j] = relu([z[b,j], zhat[b,i,j]] @ w_g1 + b_g1) @ w_g2 + b_g2
// one block per (b,j); 128 threads = H; loop over i reusing the z half
// ---------------------------------------------------------------------------
__global__ __launch_bounds__(128)
void gate_kernel(const float* __restrict__ reps, const float* __restrict__ zhat,
                 const float* __restrict__ w_g1, const float* __restrict__ b_g1,
                 const float* __restrict__ w_g2, const float* __restrict__ b_g2,
                 float* __restrict__ g) {
  __shared__ float red[4];
  int bj = blockIdx.x;
  int b = bj >> 4;
  int j = bj & 15;
  int tid = threadIdx.x;
  const float* z = reps + ((size_t)(b * NS + j) * NT) * ND;  // t = 0 row
  float hz = b_g1[tid];
  for (int d = 0; d < ND; d++) hz += z[d] * w_g1[(size_t)d * NH + tid];
  float wg2 = w_g2[tid];
  float bg2 = b_g2[0];
  for (int i = 0; i < NS; i++) {
    const float* zh = zhat + ((size_t)((b * NS + i) * NS + j)) * ND;
    float h = hz;
    for (int d = 0; d < ND; d++) h += zh[d] * w_g1[(size_t)(ND + d) * NH + tid];
    h = fmaxf(h, 0.0f);
    float p = waveSum(h * wg2);
    if ((tid & 31) == 0) red[tid >> 5] = p;
    __syncthreads();
    if (tid == 0) g[(b * NS + i) * NS + j] = red[0] + red[1] + red[2] + red[3] + bg2;
    __syncthreads();
  }
}

// ---------------------------------------------------------------------------
// K5: beta = softmax_i(g); v = [sum_i beta*zhat, z]; edge = softmax(v@w_l+b_l)
// one block per (b,j)
// ---------------------------------------------------------------------------
__global__ __launch_bounds__(256)
void edge_kernel(const float* __restrict__ reps, const float* __restrict__ zhat,
                 const float* __restrict__ g, const float* __restrict__ w_l,
                 const float* __restrict__ b_l, float* __restrict__ edge) {
  __shared__ float gb[NS];
  __shared__ float red[8][3];
  int bj = blockIdx.x;
  int b = bj >> 4;
  int j = bj & 15;
  int tid = threadIdx.x;
  int wv = tid >> 5, lane = tid & 31;
  if (tid < NS) gb[tid] = g[(b * NS + tid) * NS + j];
  __syncthreads();
  float gm = gb[0];
#pragma unroll
  for (int i = 1; i < NS; i++) gm = fmaxf(gm, gb[i]);
  float beta[NS];
  float bsum = 0.0f;
#pragma unroll
  for (int i = 0; i < NS; i++) {
    beta[i] = expf(gb[i] - gm);
    bsum += beta[i];
  }
  float inv = 1.0f / bsum;
  const float* z = reps + ((size_t)(b * NS + j) * NT) * ND;
  float p0 = 0.0f, p1 = 0.0f, p2 = 0.0f;
  for (int d = tid; d < ND; d += 256) {
    float v = 0.0f;
#pragma unroll
    for (int i = 0; i < NS; i++)
      v += beta[i] * zhat[((size_t)((b * NS + i) * NS + j)) * ND + d];
    v *= inv;
    float zd = z[d];
    p0 += v * w_l[d * 3 + 0] + zd * w_l[(ND + d) * 3 + 0];
    p1 += v * w_l[d * 3 + 1] + zd * w_l[(ND + d) * 3 + 1];
    p2 += v * w_l[d * 3 + 2] + zd * w_l[(ND + d) * 3 + 2];
  }
  p0 = waveSum(p0);
  p1 = waveSum(p1);
  p2 = waveSum(p2);
  if (lane == 0) {
    red[wv][0] = p0;
    red[wv][1] = p1;
    red[wv][2] = p2;
  }
  __syncthreads();
  if (tid == 0) {
    float lg[3];
#pragma unroll
    for (int o = 0; o < 3; o++) {
      float s = b_l[o];
#pragma unroll
      for (int w = 0; w < 8; w++) s += red[w][o];
      lg[o] = s;
    }
    float mx = fmaxf(lg[0], fmaxf(lg[1], lg[2]));
    float e0 = expf(lg[0] - mx), e1 = expf(lg[1] - mx), e2 = expf(lg[2] - mx);
    float is = 1.0f / (e0 + e1 + e2);
    edge[bj * 3 + 0] = e0 * is;
    edge[bj * 3 + 1] = e1 * is;
    edge[bj * 3 + 2] = e2 * is;
  }
}

// ---------------------------------------------------------------------------
// K6: node sim[b,s,t] = dot(cn[b,t], rn[b,s,t]) over D
// ---------------------------------------------------------------------------
__global__ __launch_bounds__(256)
void nodesim_kernel(const float* __restrict__ cn, const float* __restrict__ rn,
                    float* __restrict__ simn) {
  __shared__ float red[8];
  int blk = blockIdx.x;  // (b*16+s)*128 + t
  int b = blk >> 11;
  int t = blk & 127;
  const float* c = cn + ((size_t)(b * NT + t)) * ND;
  const float* r = rn + (size_t)blk * ND;
  float s = 0.0f;
  for (int d = threadIdx.x; d < ND; d += 256) s += c[d] * r[d];
  s = waveSum(s);
  if ((threadIdx.x & 31) == 0) red[threadIdx.x >> 5] = s;
  __syncthreads();
  if (threadIdx.x == 0) {
    float tot = 0.0f;
#pragma unroll
    for (int k = 0; k < 8; k++) tot += red[k];
    simn[blk] = tot;
  }
}

// ---------------------------------------------------------------------------
// K7: phi_r[b,s] = mean_t log(clip(T*exp_k)) @ w_r + b_r
// ---------------------------------------------------------------------------
__global__ __launch_bounds__(128)
void phir_kernel(const float* __restrict__ simn, const float* __restrict__ w_r,
                 const float* __restrict__ b_r, float* __restrict__ phir) {
  __shared__ float red[4][6];
  int bs = blockIdx.x;
  int tid = threadIdx.x;  // t
  float s = simn[(size_t)bs * NT + tid];
  float kf[6];
#pragma unroll
  for (int k = 0; k < 6; k++) {
    float d = s - c_MU[k];
    kf[k] = logf(fmaxf(128.0f * expf(d * d * c_NC[k]), 1e-6f));
  }
#pragma unroll
  for (int k = 0; k < 6; k++) {
    float v = waveSum(kf[k]);
    if ((tid & 31) == 0) red[tid >> 5][k] = v;
  }
  __syncthreads();
  if (tid == 0) {
    float out = b_r[0];
#pragma unroll
    for (int k = 0; k < 6; k++) {
      float phik = (red[0][k] + red[1][k] + red[2][k] + red[3][k]) / 128.0f;
      out += phik * w_r[k];
    }
    phir[bs] = out;
  }
}

// ---------------------------------------------------------------------------
// K8: rationale = softmax_s(phi_r); out[b,o] = sum_s edge[b,s,o]*rationale[b,s]
// ---------------------------------------------------------------------------
__global__ __launch_bounds__(32)
void final_kernel(const float* __restrict__ edge, const float* __restrict__ phir,
                  float* __restrict__ out) {
  int b = blockIdx.x;
  int tid = threadIdx.x;
  float m = -3.0e38f;
#pragma unroll
  for (int s = 0; s < NS; s++) m = fmaxf(m, phir[b * NS + s]);
  float rat[NS];
  float sum = 0.0f;
#pragma unroll
  for (int s = 0; s < NS; s++) {
    rat[s] = expf(phir[b * NS + s] - m);
    sum += rat[s];
  }
  float inv = 1.0f / sum;
  if (tid < 3) {
    float o = 0.0f;
#pragma unroll
    for (int s = 0; s < NS; s++) o += edge[(b * NS + s) * 3 + tid] * rat[s] * inv;
    out[b * 3 + tid] = o;
  }
}

// ---------------------------------------------------------------------------
extern "C" void kernel_launch(void* const* d_in, const int* in_sizes, int n_in,
                              void* d_out, int out_size, void* d_ws, size_t ws_size,
                              hipStream_t stream) {
  const float* claim_reps = (const float*)d_in[0];   // (B,T,D)
  const float* reps = (const float*)d_in[1];         // (B,S,T,D)
  // d_in[2] claim_token_mask (unused by reference), d_in[3] token_mask (all ones)
  const float* w_sel = (const float*)d_in[4];
  const float* b_sel = (const float*)d_in[5];
  const float* w_g1 = (const float*)d_in[6];
  const float* b_g1 = (const float*)d_in[7];
  const float* w_g2 = (const float*)d_in[8];
  const float* b_g2 = (const float*)d_in[9];
  const float* w_r = (const float*)d_in[10];
  const float* b_r = (const float*)d_in[11];
  const float* w_l = (const float*)d_in[12];
  const float* b_l = (const float*)d_in[13];
  float* out = (float*)d_out;

  // workspace layout (floats)
  float* rn = (float*)d_ws;                               // B*S*T*D = 6291456
  float* cn = rn + (size_t)NB * NS * NT * ND;             // B*T*D   = 393216
  float* att = cn + (size_t)NB * NT * ND;                 // B*S*S*T = 131072
  float* zhat = att + (size_t)NB * NS * NS * NT;          // B*S*S*D = 786432
  float* g = zhat + (size_t)NB * NS * NS * ND;            // B*S*S   = 1024
  float* edge = g + (size_t)NB * NS * NS;                 // B*S*3   = 192
  float* simn = edge + (size_t)NB * NS * 3;               // B*S*T   = 8192
  float* phir = simn + (size_t)NB * NS * NT;              // B*S     = 64

  norm_kernel<<<dim3(NB * NS * NT + NB * NT), dim3(256), 0, stream>>>(reps, claim_reps, rn, cn);
  pool_att_kernel<<<dim3(NB * NS * NS), dim3(256), 0, stream>>>(rn, w_sel, b_sel, att);
  zhat_kernel<<<dim3(NB * NS * NS), dim3(256), 0, stream>>>(reps, att, zhat);
  gate_kernel<<<dim3(NB * NS), dim3(128), 0, stream>>>(reps, zhat, w_g1, b_g1, w_g2, b_g2, g);
  edge_kernel<<<dim3(NB * NS), dim3(256), 0, stream>>>(reps, zhat, g, w_l, b_l, edge);
  nodesim_kernel<<<dim3(NB * NS * NT), dim3(256), 0, stream>>>(cn, rn, simn);
  phir_kernel<<<dim3(NB * NS), dim3(128), 0, stream>>>(simn, w_r, b_r, phir);
  final_kernel<<<dim3(NB), dim3(32), 0, stream>>>(edge, phir, out);
}